// ProgressiveFeatureAdjustment_82901458747713
// MI455X (gfx1250) — compile-verified
//
#include <hip/hip_runtime.h>
#include <stdint.h>

#define N_PTS 65536
#define DIMS  256
#define NCLS  200
#define CT    13          // ceil(200/16) column tiles
#define KC    64          // K-chunk staged in LDS
#define KSTEPS (KC / 4)   // 16 wmma K-steps per chunk

constexpr float TEMP_F  = 0.07f;
constexpr float MOM     = 0.99f;
constexpr float CONFID_F= 0.95f;
constexpr float EPS_F   = 1e-12f;

typedef __attribute__((ext_vector_type(2))) float v2f;
typedef __attribute__((ext_vector_type(8))) float v8f;

// ---------------------------------------------------------------------------
// Kernel A: per-row inverse L2 norms for q_c and k_c (one wave per row).
// ---------------------------------------------------------------------------
__global__ __launch_bounds__(256) void row_inv_norm_kernel(
    const float* __restrict__ q, const float* __restrict__ k,
    float* __restrict__ invq, float* __restrict__ invk) {
  int wave = (blockIdx.x * blockDim.x + threadIdx.x) >> 5;   // global wave id
  int lane = threadIdx.x & 31;
  const float* src;
  float* dst;
  int row;
  if (wave < N_PTS) { src = q; dst = invq; row = wave; }
  else              { src = k; dst = invk; row = wave - N_PTS; }
  const float4* p = (const float4*)(src + (size_t)row * DIMS);
  float4 a = p[lane];        // 32 lanes * 2 float4 = 256 floats
  float4 b = p[lane + 32];
  float s = a.x*a.x + a.y*a.y + a.z*a.z + a.w*a.w
          + b.x*b.x + b.y*b.y + b.z*b.z + b.w*b.w;
  #pragma unroll
  for (int off = 16; off > 0; off >>= 1) s += __shfl_xor(s, off, 32);
  if (lane == 0) dst[row] = 1.0f / fmaxf(sqrtf(s), EPS_F);
}

// ---------------------------------------------------------------------------
// Kernel B: logits = (qn @ queue_list) / TEMP with V_WMMA_F32_16X16X4_F32.
// B is staged per-block into LDS in WMMA per-lane pair layout using the
// gfx1250 async global->LDS engine (global_load_async_to_lds_b32, ASYNCcnt).
// Inner loop: 1 global b64 (A) + 13 conflict-free ds_load_b64 + 13 WMMA.
// ---------------------------------------------------------------------------
__global__ __launch_bounds__(256) void logits_wmma_kernel(
    const float* __restrict__ q, const float* __restrict__ proto,
    const float* __restrict__ invq, float* __restrict__ out) {
  __shared__ float2 bstage[KSTEPS * CT * 32];      // 16*13*32*8 = 53,248 B

  const int tid   = threadIdx.x;
  const int lane  = tid & 31;
  const int w     = tid >> 5;
  const int mtile = blockIdx.x * 8 + w;            // 4096 tiles total
  const int hh    = lane >> 4;                     // 0: lanes 0-15, 1: 16-31
  const int lx    = lane & 15;
  const int koff  = hh * 2;                        // K offset held by this half

  // Zero the permanently-invalid pad slots (tile 12, cols 200..207) once;
  // async staging never writes them, so they stay zero for every chunk.
  {
    const int zk = tid >> 4;                       // 0..15 kstep
    const int j  = tid & 15;
    const int zl = (j < 8) ? (8 + j) : (16 + j);   // lanes 8-15, 24-31
    bstage[(zk * CT + 12) * 32 + zl] = make_float2(0.0f, 0.0f);
  }

  const uint32_t lbase = (uint32_t)(uintptr_t)(void*)&bstage[0]; // LDS byte offset

  v8f acc[CT];
  #pragma unroll
  for (int t = 0; t < CT; ++t)
    #pragma unroll
    for (int r = 0; r < 8; ++r) acc[t][r] = 0.0f;

  const float* arow = q + (size_t)(mtile * 16 + lx) * DIMS;

  for (int chunk = 0; chunk < DIMS / KC; ++chunk) {
    const int kbase = chunk * KC;

    // ---- stage B[kbase..kbase+63][*] as wmma-ordered float2 pairs --------
    // Whole wave handles one (kstep, tile) per iteration; lane == own lane.
    #pragma unroll 1
    for (int i = 0; i < (KSTEPS * CT) / 8; ++i) {  // 26 iterations
      const int flat  = w + i * 8;                 // 0..207
      const int kstep = flat / CT;
      const int t     = flat - kstep * CT;
      const int n     = t * 16 + lx;
      if (n < NCLS) {
        const int krow = kbase + kstep * 4 + koff;
        uint32_t g0 = (uint32_t)((krow * NCLS + n) * 4);
        uint32_t g1 = g0 + NCLS * 4;
        uint32_t l0 = lbase + (uint32_t)(((kstep * CT + t) * 32 + lane) * 8);
        uint32_t l1 = l0 + 4;
        asm volatile("global_load_async_to_lds_b32 %0, %1, %2"
                     :: "v"(l0), "v"(g0), "s"(proto) : "memory");
        asm volatile("global_load_async_to_lds_b32 %0, %1, %2"
                     :: "v"(l1), "v"(g1), "s"(proto) : "memory");
      }
    }
    asm volatile("s_wait_asynccnt 0x0" ::: "memory");
    __syncthreads();

    // ---- compute: 16 K-steps on the staged chunk -------------------------
    #pragma unroll 4
    for (int kstep = 0; kstep < KSTEPS; ++kstep) {
      v2f a;
      const float* ap = arow + kbase + kstep * 4 + koff;
      a.x = ap[0];
      a.y = ap[1];
      #pragma unroll
      for (int t = 0; t < CT; ++t) {
        float2 bb = bstage[(kstep * CT + t) * 32 + lane];
        v2f b; b.x = bb.x; b.y = bb.y;
        acc[t] = __builtin_amdgcn_wmma_f32_16x16x4_f32(
            /*neg_a=*/false, a, /*neg_b=*/false, b,
            /*c_mod=*/(short)0, acc[t], /*reuse_a=*/false, /*reuse_b=*/false);
      }
    }
    __syncthreads();   // all waves done reading before next chunk overwrites
  }

  // Output scale: inv_norm(q_row) / TEMP  (row depends on r + half, not t)
  float sc[8];
  #pragma unroll
  for (int r = 0; r < 8; ++r) {
    const int row = mtile * 16 + r + hh * 8;
    sc[r] = invq[row] * (1.0f / TEMP_F);
  }
  #pragma unroll
  for (int t = 0; t < CT; ++t) {
    const int n = t * 16 + lx;
    if (n < NCLS) {
      #pragma unroll
      for (int r = 0; r < 8; ++r) {
        const int row = mtile * 16 + r + hh * 8;
        out[(size_t)row * NCLS + n] = acc[t][r] * sc[r];
      }
    }
  }
}

// ---------------------------------------------------------------------------
// Kernel C: ordered momentum scan per class. One block per class, thread d
// owns dimension d. Per 256-sample chunk: wave32 ballots -> LDS -> every
// thread replays the ordered bit-scan on its own column element.
// ---------------------------------------------------------------------------
__global__ __launch_bounds__(256) void queue_update_kernel(
    const float* __restrict__ kc,    const float* __restrict__ proto,
    const float* __restrict__ confid,const int*   __restrict__ labels,
    const int*   __restrict__ pivot_in, const float* __restrict__ invk,
    float* __restrict__ queue_out,   int* __restrict__ pivot_out) {
  const int c    = blockIdx.x;
  const int d    = threadIdx.x;          // 0..255
  const int wid  = threadIdx.x >> 5;
  const int lane = threadIdx.x & 31;

  __shared__ unsigned smask[8];

  float col = proto[(size_t)d * NCLS + c];
  const int p0 = pivot_in[c];
  bool did_first = false;
  bool any = false;

  for (int base = 0; base < N_PTS; base += 256) {
    const int i = base + threadIdx.x;
    const bool match = (labels[i] == c) && (confid[i] >= CONFID_F);
    const unsigned m = (unsigned)__ballot(match);   // wave32: low 32 bits
    if (lane == 0) smask[wid] = m;
    __syncthreads();
    unsigned mw[8];
    #pragma unroll
    for (int ww = 0; ww < 8; ++ww) mw[ww] = smask[ww];
    __syncthreads();
    #pragma unroll
    for (int ww = 0; ww < 8; ++ww) {
      unsigned mask = mw[ww];
      while (mask) {
        const int b = __ffs(mask) - 1;
        mask &= mask - 1;
        const int idx = base + ww * 32 + b;         // ordered selected sample
        const float kn = kc[(size_t)idx * DIMS + d] * invk[idx];
        if (p0 == 0 && !did_first) { col = kn; did_first = true; }
        else                       { col = col * MOM + kn * (1.0f - MOM); }
        any = true;
      }
    }
  }

  queue_out[(size_t)d * NCLS + c] = col;
  if (threadIdx.x == 0) pivot_out[c] = p0 + ((p0 == 0 && any) ? 1 : 0);
}

// ---------------------------------------------------------------------------
extern "C" void kernel_launch(void* const* d_in, const int* in_sizes, int n_in,
                              void* d_out, int out_size, void* d_ws, size_t ws_size,
                              hipStream_t stream) {
  const float* q_c        = (const float*)d_in[0];
  const float* k_c        = (const float*)d_in[1];
  const float* queue_list = (const float*)d_in[2];
  const float* confid_q   = (const float*)d_in[3];
  const int*   labels     = (const int*)  d_in[4];
  const int*   queue_piv  = (const int*)  d_in[5];

  float* out       = (float*)d_out;
  float* logits    = out;                                         // N*C
  float* new_queue = out + (size_t)N_PTS * NCLS;                  // D*C
  int*   new_pivot = (int*)(out + (size_t)N_PTS * NCLS + (size_t)DIMS * NCLS);

  float* invq = (float*)d_ws;        // N floats
  float* invk = invq + N_PTS;        // N floats

  // A: 2*N rows, 8 rows (waves) per block
  row_inv_norm_kernel<<<(2 * N_PTS) / 8, 256, 0, stream>>>(q_c, k_c, invq, invk);
  // B: 4096 M-tiles, 8 waves per block
  logits_wmma_kernel<<<(N_PTS / 16) / 8, 256, 0, stream>>>(q_c, queue_list, invq, logits);
  // C: one block per class
  queue_update_kernel<<<NCLS, 256, 0, stream>>>(k_c, queue_list, confid_q, labels,
                                                queue_piv, invk, new_queue, new_pivot);
}